// Repeat_Recommendation_Decoder_12146167513170
// MI455X (gfx1250) — compile-verified
//
#include <hip/hip_runtime.h>
#include <hip/hip_bf16.h>
#include <math.h>

// Problem constants (match reference)
#define BB 1024
#define SS 200
#define HH 128
#define NN 100000
#define BS (BB * SS) // 204800

#define LDS_STRIDE 132   // 128 + 4 pad floats: conflict-free ds_load_b64 pattern

typedef float v2f __attribute__((ext_vector_type(2)));
typedef float v8f __attribute__((ext_vector_type(8)));

// ---------------------------------------------------------------------------
// CDNA5 async global->LDS copy of one 16-byte chunk (ASYNCcnt-tracked).
// LDS operand is the wave-relative byte offset (= low 32 bits of the generic
// pointer to a __shared__ object on gfx1250).
// ---------------------------------------------------------------------------
__device__ __forceinline__ void async_copy_b128(const float* gsrc, float* ldst) {
  asm volatile("global_load_async_to_lds_b128 %0, %1, off"
               :: "v"((unsigned)(unsigned long long)ldst), "v"(gsrc)
               : "memory");
}
__device__ __forceinline__ void wait_asynccnt0() {
  asm volatile("s_wait_asynccnt 0x0" ::: "memory");
}

// ---------------------------------------------------------------------------
// Unified 128-row x 128-col fp32 GEMM block:  D = X[128 rows] @ W^T
//   - A panel (64KB) and W (64KB) async-copied to LDS (stride-132 padded so
//     the ds_load_b64 fragment reads hit all 64 banks without conflict)
//   - per wave: 32-step K loop, 8x v_wmma_f32_16x16x4_f32 per step,
//     register double buffering (prefetch hb+1 fragments during hb's WMMAs)
//
// A 16x4 f32 layout (ISA 7.12.2): lane L holds row M=L%16;
//   VGPR0 = A[M][koff], VGPR1 = A[M][koff+1], koff = (L<16) ? 0 : 2.
// B mirrored; B[k][n] = W[n][k] is contiguous in h.
// D layout: vgpr v, half -> M = v + 8*half, N = lane%16.
//
// FUSED=false: store raw D to outD[m*H+n]            (lm projection)
// FUSED=true : scores[row] = Vr . tanh(D + lm_proj[b]) + Vr_b, masked
// ---------------------------------------------------------------------------
template <bool FUSED>
__global__ void gemm128_kernel(const float* __restrict__ X,
                               const float* __restrict__ W,
                               float* __restrict__ outD,
                               const float* __restrict__ lm_proj,
                               const float* __restrict__ Vr_w,
                               const float* __restrict__ Vr_b,
                               const unsigned char* __restrict__ mask,
                               float* __restrict__ scores) {
  extern __shared__ float smem[];
  float* sA = smem;                        // [128][132]
  float* sW = smem + 128 * LDS_STRIDE;     // [128][132]

  const int t    = threadIdx.x;
  const int lane = t & 31;
  const int wv   = t >> 5;                 // wave in block: 0..7
  const int half = lane >> 4;
  const int l16  = lane & 15;
  const int koff = half * 2;
  const int blockRowBase = blockIdx.x * 128;

  // ---- async stage: A panel + W, coalesced b128 per lane ----
#pragma unroll
  for (int j = t; j < 128 * 32; j += 256) {   // 4096 x float4 per surface
    const int row = j >> 5;
    const int col = (j & 31) << 2;
    async_copy_b128(X + (size_t)(blockRowBase + row) * HH + col,
                    sA + row * LDS_STRIDE + col);
    async_copy_b128(W + (size_t)row * HH + col,
                    sW + row * LDS_STRIDE + col);
  }
  wait_asynccnt0();
  __syncthreads();

  // ---- K loop: LDS-fed WMMA with register double buffering ----
  v8f c[8];
#pragma unroll
  for (int nt = 0; nt < 8; ++nt) c[nt] = (v8f)0.f;

  const float* aBase = sA + (wv * 16 + l16) * LDS_STRIDE + koff;

  v2f aBuf[2];
  v2f bBuf[2][8];
  aBuf[0] = *reinterpret_cast<const v2f*>(aBase);
#pragma unroll
  for (int nt = 0; nt < 8; ++nt)
    bBuf[0][nt] =
        *reinterpret_cast<const v2f*>(sW + (nt * 16 + l16) * LDS_STRIDE + koff);

#pragma unroll 4
  for (int hb = 0; hb < 32; ++hb) {
    const int cur = hb & 1;
    const int nxt = cur ^ 1;
    const int hN  = ((hb + 1) & 31) * 4;   // wraps on last iter (harmless)
    // prefetch next fragments from LDS while WMMAs below execute
    aBuf[nxt] = *reinterpret_cast<const v2f*>(aBase + hN);
#pragma unroll
    for (int nt = 0; nt < 8; ++nt)
      bBuf[nxt][nt] = *reinterpret_cast<const v2f*>(
          sW + (nt * 16 + l16) * LDS_STRIDE + hN + koff);
#pragma unroll
    for (int nt = 0; nt < 8; ++nt)
      c[nt] = __builtin_amdgcn_wmma_f32_16x16x4_f32(
          false, aBuf[cur], false, bBuf[cur][nt], (short)0, c[nt], false, false);
  }

  const int rowbase = blockRowBase + wv * 16;

  if (!FUSED) {
    // ---- raw D store (lm projection) ----
#pragma unroll
    for (int nt = 0; nt < 8; ++nt) {
#pragma unroll
      for (int v = 0; v < 8; ++v) {
        const int m = rowbase + v + 8 * half;
        const int n = nt * 16 + l16;
        outD[(size_t)m * HH + n] = c[nt][v];
      }
    }
    return;
  }

  // ---- fused epilogue: +lm_proj, tanh, x Vr, 16-lane butterfly reduce ----
  v8f acc = (v8f)0.f;
#pragma unroll
  for (int nt = 0; nt < 8; ++nt) {
    const int n = nt * 16 + l16;
    const float vr = Vr_w[n];
#pragma unroll
    for (int v = 0; v < 8; ++v) {
      const int row = rowbase + v + 8 * half;
      const int b   = row / SS;
      const float lmv = lm_proj[(size_t)b * HH + n];
      acc[v] += tanhf(c[nt][v] + lmv) * vr;
    }
  }

  const float bias = Vr_b[0];
#pragma unroll
  for (int v = 0; v < 8; ++v) {
    float s = acc[v];
    s += __shfl_xor(s, 1, 32);
    s += __shfl_xor(s, 2, 32);
    s += __shfl_xor(s, 4, 32);
    s += __shfl_xor(s, 8, 32);
    if (l16 == 0) {
      const int row = rowbase + v + 8 * half;
      float sc = s + bias;
      if (mask[row]) sc = -1e9f;
      scores[row] = sc;
    }
  }
}

// ---------------------------------------------------------------------------
// K3: in-place softmax over S=200 per batch row. One block per row.
// ---------------------------------------------------------------------------
__global__ void softmax_kernel(float* __restrict__ scores) {
  const int b = blockIdx.x;
  const int t = threadIdx.x;
  __shared__ float red[256];

  float x = (t < SS) ? scores[(size_t)b * SS + t] : -INFINITY;
  red[t] = x;
  __syncthreads();
#pragma unroll
  for (int o = 128; o > 0; o >>= 1) {
    if (t < o) red[t] = fmaxf(red[t], red[t + o]);
    __syncthreads();
  }
  const float mx = red[0];
  __syncthreads();

  float e = (t < SS) ? __expf(x - mx) : 0.f;
  red[t] = e;
  __syncthreads();
#pragma unroll
  for (int o = 128; o > 0; o >>= 1) {
    if (t < o) red[t] += red[t + o];
    __syncthreads();
  }
  const float inv = 1.f / red[0];
  if (t < SS) scores[(size_t)b * SS + t] = e * inv;
}

// ---------------------------------------------------------------------------
// K4: scatter-add att into zeroed [B, N] output (global_atomic_add_f32).
// ---------------------------------------------------------------------------
__global__ void scatter_kernel(const int* __restrict__ item_seq,
                               const float* __restrict__ att,
                               float* __restrict__ out) {
  const int i = blockIdx.x * blockDim.x + threadIdx.x;
  if (i < BS) {
    const int b = i / SS;
    atomicAdd(out + (size_t)b * NN + item_seq[i], att[i]);
  }
}

extern "C" void kernel_launch(void* const* d_in, const int* in_sizes, int n_in,
                              void* d_out, int out_size, void* d_ws, size_t ws_size,
                              hipStream_t stream) {
  const float*         all_memory  = (const float*)d_in[0];         // [B,S,H]
  const float*         last_memory = (const float*)d_in[1];         // [B,H]
  const int*           item_seq    = (const int*)d_in[2];           // [B,S]
  const unsigned char* mask        = (const unsigned char*)d_in[3]; // [B,S] bool
  const float*         Ur_w        = (const float*)d_in[4];         // [H,H]
  const float*         Wr_w        = (const float*)d_in[5];         // [H,H]
  const float*         Vr_w        = (const float*)d_in[6];         // [1,H]
  const float*         Vr_b        = (const float*)d_in[7];         // [1]
  float* out = (float*)d_out;                                       // [B,N]

  // Workspace layout: lm_proj [B,H] | scores/att [B,S]
  float* lm_proj = (float*)d_ws;
  float* scores  = lm_proj + (size_t)BB * HH;

  const size_t shmem = (size_t)2 * 128 * LDS_STRIDE * sizeof(float); // 132 KiB
  (void)hipFuncSetAttribute((const void*)gemm128_kernel<false>,
                            hipFuncAttributeMaxDynamicSharedMemorySize,
                            (int)shmem);
  (void)hipFuncSetAttribute((const void*)gemm128_kernel<true>,
                            hipFuncAttributeMaxDynamicSharedMemorySize,
                            (int)shmem);

  // K1: lm projection (1024 rows -> 8 blocks)
  gemm128_kernel<false><<<dim3(BB / 128), dim3(256), shmem, stream>>>(
      last_memory, Wr_w, lm_proj, nullptr, nullptr, nullptr, nullptr, nullptr);

  // K2: fused GEMM + tanh + Vr reduction -> scores (204800 rows -> 1600 blocks)
  gemm128_kernel<true><<<dim3(BS / 128), dim3(256), shmem, stream>>>(
      all_memory, Ur_w, nullptr, lm_proj, Vr_w, Vr_b, mask, scores);

  // K3: softmax per row (in place -> att)
  softmax_kernel<<<dim3(BB), dim3(256), 0, stream>>>(scores);

  // Zero the 410 MB output at full bandwidth, then scatter-add.
  hipMemsetAsync(d_out, 0, (size_t)out_size * sizeof(float), stream);
  scatter_kernel<<<dim3((BS + 255) / 256), dim3(256), 0, stream>>>(
      item_seq, scores, out);
}